// Prototypes_37950331027884
// MI455X (gfx1250) — compile-verified
//
#include <hip/hip_runtime.h>
#include <hip/hip_bf16.h>

#define BATCH 4
#define NCLS  19
#define DIM   512
#define HV    128
#define WV    256
#define NPIX  (BATCH*HV*WV)   // 131072
#define HS    512
#define WSG   1024
#define HC    64
#define WC    128
#define TOPK  32
#define TA    128             // threads per scan block
#define SPLITS 16             // pixel chunks per class -> 19*16 = 304 blocks
#define CHUNK (NPIX/SPLITS)   // 8192
#define NTILE (DIM/16)        // 32 dim tiles

typedef __attribute__((ext_vector_type(2))) float v2f;
typedef __attribute__((ext_vector_type(8))) float v8f;

static __device__ __forceinline__ unsigned long long umax64(unsigned long long a,
                                                            unsigned long long b) {
  return a > b ? a : b;
}

// ---------------------------------------------------------------------------
// Kernel A: grid (SPLITS, NCLS). Each block scans an 8192-pixel chunk for its
// class: fused nearest-label/mask + bilinear-CAM weight, per-thread local
// top-32 (LDS-resident), then a 32-round tree-max merge. Emits the chunk's
// top-32 packed keys ([ordered f32 weight | ~pixel-idx] u64 -> max order ==
// lax.top_k value order with smallest-index tie-break). Exact for any
// label distribution since global top-32 keys survive chunk-local top-32.
// ---------------------------------------------------------------------------
__global__ void proto_topk_kernel(const float* __restrict__ cam,
                                  const int*   __restrict__ seg,
                                  const int*   __restrict__ dmask,
                                  unsigned long long* __restrict__ cand_g) {
  __shared__ unsigned long long cand[TA * TOPK];   // 32 KB
  __shared__ unsigned long long pmax[TA];          // 1 KB
  const int sp = blockIdx.x;
  const int c  = blockIdx.y;
  const int t  = threadIdx.x;

  unsigned long long* my = &cand[t * TOPK];
#pragma unroll
  for (int s = 0; s < TOPK; ++s) my[s] = 0ull;

  int cnt = 0;
  const int base = sp * CHUNK;
  for (int n = base + t; n < base + CHUNK; n += TA) {
    const int j = n & (WV - 1);
    const int i = (n >> 8) & (HV - 1);
    const int b = n >> 15;
    // nearest-neighbor downsample (512x1024 -> 128x256): source stride 4
    const int sidx = b * (HS * WSG) + (i * 4) * WSG + (j * 4);
    // prefetch a few strided iterations ahead (gfx1250 global_prefetch_b8)
    if (n + 4 * TA < base + CHUNK) {
      const int nn = n + 4 * TA;
      const int pj = nn & (WV - 1), pi = (nn >> 8) & (HV - 1), pb = nn >> 15;
      __builtin_prefetch(&seg[pb * (HS * WSG) + (pi * 4) * WSG + (pj * 4)], 0, 0);
    }
    const int lab = seg[sidx];
    const int dm  = dmask[sidx];
    if ((lab != c) | (dm != 1)) continue;   // also rejects IGNORE=255
    // bilinear align_corners=True upsample of cam_map (64x128 -> 128x256)
    const float ph = (float)i * (63.0f / 127.0f);
    const float pw = (float)j * (127.0f / 255.0f);
    const int h0 = (int)ph; const float fh = ph - (float)h0;
    const int w0 = (int)pw; const float fw = pw - (float)w0;
    const int h1 = min(h0 + 1, HC - 1);
    const int w1 = min(w0 + 1, WC - 1);
    const float* cb = cam + (size_t)(b * NCLS + c) * (HC * WC);
    const float v00 = cb[h0 * WC + w0], v01 = cb[h0 * WC + w1];
    const float v10 = cb[h1 * WC + w0], v11 = cb[h1 * WC + w1];
    const float xr0 = v00 * (1.f - fh) + v10 * fh;
    const float xr1 = v01 * (1.f - fh) + v11 * fh;
    const float wv  = xr0 * (1.f - fw) + xr1 * fw;   // in [0,1] -> non-negative
    const unsigned wbits = __float_as_uint(wv) | 0x80000000u; // ordered, >0
    const unsigned long long key =
        ((unsigned long long)wbits << 32) |
        (unsigned long long)(0xFFFFFFFFu - (unsigned)n);
    if (cnt < TOPK) {
      my[cnt++] = key;
    } else {
      int mp = 0; unsigned long long mv = my[0];
#pragma unroll
      for (int s = 1; s < TOPK; ++s) {
        unsigned long long q = my[s];
        if (q < mv) { mv = q; mp = s; }
      }
      if (key > mv) my[mp] = key;
    }
  }
  __syncthreads();

  for (int r = 0; r < TOPK; ++r) {
    unsigned long long m = 0ull;
#pragma unroll
    for (int s = 0; s < TOPK; ++s) m = umax64(m, my[s]);
    pmax[t] = m;
    __syncthreads();
    for (int s = TA / 2; s > 0; s >>= 1) {
      if (t < s) pmax[t] = umax64(pmax[t], pmax[t + s]);
      __syncthreads();
    }
    const unsigned long long win = pmax[0];
    __syncthreads();                 // all threads consumed pmax[0]
    if (win != 0ull) {               // unique owner invalidates its slot
#pragma unroll
      for (int s = 0; s < TOPK; ++s) {
        if (my[s] == win) { my[s] = 0ull; break; }
      }
    }
    if (t == 0) cand_g[(c * SPLITS + sp) * TOPK + r] = win;
  }
}

// -------- helpers: statically-register-allocated tile load / WMMA chain ----
static __device__ __forceinline__ void load_tile(const float* __restrict__ v,
                                                 const int (&ofA)[8],
                                                 const int (&ofB)[8],
                                                 int tile, int ln,
                                                 float (&bx)[8], float (&by)[8]) {
  const size_t dd = (size_t)(tile * 16 + ln) * (HV * WV);
#pragma unroll
  for (int kc = 0; kc < 8; ++kc) {
    bx[kc] = v[(size_t)ofA[kc] + dd];
    by[kc] = v[(size_t)ofB[kc] + dd];
  }
}

static __device__ __forceinline__ void compute_tile(const v2f (&af)[8],
                                                    const float (&bx)[8],
                                                    const float (&by)[8],
                                                    float* agg, int tile,
                                                    int half, int ln) {
  v8f acc = {0.f, 0.f, 0.f, 0.f, 0.f, 0.f, 0.f, 0.f};
#pragma unroll
  for (int kc = 0; kc < 8; ++kc) {
    v2f bb; bb.x = bx[kc]; bb.y = by[kc];
    acc = __builtin_amdgcn_wmma_f32_16x16x4_f32(
        /*neg_a=*/false, af[kc], /*neg_b=*/false, bb,
        /*c_mod=*/(short)0, acc, /*reuse_a=*/false, /*reuse_b=*/false);
  }
  if (half == 0) agg[tile * 16 + ln] = acc[0];     // D: VGPR0 lanes0-15 = M0
}

// ---------------------------------------------------------------------------
// Kernel B: one wave32 per class.
//  Stage 1: merge SPLITS*32 = 512 candidate keys held in registers (16/lane);
//           32 rounds of lane-local max + __shfl_xor wave max; winner r is
//           decoded by all lanes (uniform wsum/nsel) and parked in LDS.
//  Stage 2: agg[1,512] = w[1,32] x F[32,512] on V_WMMA_F32_16X16X4_F32.
//           A-fragments + per-lane gather offsets hoisted to registers;
//           feature loads double-buffered in two statically-named register
//           sets (two tiles per iteration -> no movrel indexing).
//  Stage 3: normalize / EMA / renormalize; write pnew and new_count.
// ---------------------------------------------------------------------------
__global__ void proto_agg_kernel(const float* __restrict__ v,
                                 const float* __restrict__ proto,
                                 const float* __restrict__ ucount,
                                 const unsigned long long* __restrict__ cand_g,
                                 float* __restrict__ out) {
  const int c = blockIdx.x;
  const int l = threadIdx.x;           // 0..31, one full wave
  __shared__ float wsh[TOPK];
  __shared__ int   offsh[TOPK];
  __shared__ float agg[DIM];
  __shared__ float red[32];

  // ---- Stage 1: wave-local merge of 512 keys ----
  unsigned long long k[SPLITS];
#pragma unroll
  for (int q = 0; q < SPLITS; ++q)
    k[q] = cand_g[(c * SPLITS + q) * TOPK + l];

  float wsum = 0.f; float nsel = 0.f;
  for (int r = 0; r < TOPK; ++r) {
    unsigned long long m = k[0];
#pragma unroll
    for (int q = 1; q < SPLITS; ++q) m = umax64(m, k[q]);
#pragma unroll
    for (int off = 16; off > 0; off >>= 1)
      m = umax64(m, (unsigned long long)__shfl_xor((unsigned long long)m, off, 32));
    // m uniform across the wave now
    bool owned = false;
#pragma unroll
    for (int q = 0; q < SPLITS; ++q) {
      if (!owned && m != 0ull && k[q] == m) { k[q] = 0ull; owned = true; }
    }
    if (m != 0ull) {
      const float    wv = __uint_as_float((unsigned)(m >> 32) & 0x7FFFFFFFu);
      const unsigned n  = 0xFFFFFFFFu - (unsigned)(m & 0xFFFFFFFFull);
      wsum += wv; nsel += 1.f;                       // uniform
      if (l == r) {
        const int j = (int)(n & (WV - 1));
        const int i = (int)((n >> 8) & (HV - 1));
        const int b = (int)(n >> 15);
        wsh[r]   = wv;
        offsh[r] = b * (DIM * HV * WV) + i * WV + j; // + d*HV*WV per dim
      }
    } else if (l == r) {
      wsh[r] = 0.f;                                  // isfinite==false -> 0
      offsh[r] = 0;                                  // safe gather index
    }
  }
  __syncthreads();

  const int half = l >> 4;   // 0: lanes 0-15, 1: lanes 16-31
  const int ln   = l & 15;

  // ---- Stage 2: WMMA aggregation ----
  // Hoisted A-fragments: row M=0 only; lane0 carries K0,K1 / lane16 K2,K3.
  v2f af[8];
  int ofA[8], ofB[8];
#pragma unroll
  for (int kc = 0; kc < 8; ++kc) {
    const int k0 = kc * 4 + half * 2;
    const float a0 = wsh[k0], a1 = wsh[k0 + 1];
    af[kc].x = (ln == 0) ? a0 : 0.f;
    af[kc].y = (ln == 0) ? a1 : 0.f;
    ofA[kc] = offsh[k0];
    ofB[kc] = offsh[k0 + 1];
  }

  float b0x[8], b0y[8], b1x[8], b1y[8];
  load_tile(v, ofA, ofB, 0, ln, b0x, b0y);
  for (int tile = 0; tile < NTILE; tile += 2) {
    load_tile(v, ofA, ofB, tile + 1, ln, b1x, b1y);     // prefetch odd tile
    compute_tile(af, b0x, b0y, agg, tile, half, ln);    // consume even tile
    if (tile + 2 < NTILE)
      load_tile(v, ofA, ofB, tile + 2, ln, b0x, b0y);   // prefetch next even
    compute_tile(af, b1x, b1y, agg, tile + 1, half, ln);// consume odd tile
  }
  __syncthreads();

  // ---- Stage 3: normalize / EMA / renormalize ----
  const float div = (wsum == 0.f) ? 1.f : wsum;
  float av[16]; float ss = 0.f;
#pragma unroll
  for (int q = 0; q < 16; ++q) {
    const float x = agg[l * 16 + q] / div;
    av[q] = x; ss += x * x;
  }
  red[l] = ss; __syncthreads();
  if (l == 0) { float s = 0.f; for (int q = 0; q < 32; ++q) s += red[q]; red[0] = s; }
  __syncthreads();
  ss = red[0];
  __syncthreads();
  const float inv = 1.0f / sqrtf(fmaxf(ss, 1e-24f));

  const float uc = ucount[c];
  const bool present = nsel > 0.f;
  const float alpha  = (uc == 0.f) ? 1.f : fminf(1.f - 1.f / (uc + 1.f), 0.001f);
  const bool do_upd  = present && (wsum != 0.f);

  float pv[16]; float ss2 = 0.f;
#pragma unroll
  for (int q = 0; q < 16; ++q) {
    const float p  = proto[c * DIM + l * 16 + q];
    const float pn = do_upd ? (p * (1.f - alpha) + alpha * (av[q] * inv)) : p;
    pv[q] = pn; ss2 += pn * pn;
  }
  red[l] = ss2; __syncthreads();
  if (l == 0) { float s = 0.f; for (int q = 0; q < 32; ++q) s += red[q]; red[0] = s; }
  __syncthreads();
  const float inv2 = 1.0f / sqrtf(fmaxf(red[0], 1e-24f));
#pragma unroll
  for (int q = 0; q < 16; ++q)
    out[c * DIM + l * 16 + q] = pv[q] * inv2;
  if (l == 0)
    out[NCLS * DIM + c] = uc + (present ? nsel : 0.f);
}

// ---------------------------------------------------------------------------
extern "C" void kernel_launch(void* const* d_in, const int* in_sizes, int n_in,
                              void* d_out, int out_size, void* d_ws, size_t ws_size,
                              hipStream_t stream) {
  const float* v      = (const float*)d_in[0];   // [4,512,128,256]
  const float* cam    = (const float*)d_in[1];   // [4,19,64,128]
  const int*   seg    = (const int*)  d_in[2];   // [4,1,512,1024]
  const int*   dmask  = (const int*)  d_in[3];   // [4,1,512,1024]
  const float* proto  = (const float*)d_in[4];   // [19,512]
  const float* ucount = (const float*)d_in[5];   // [19]
  float* out = (float*)d_out;                    // [19*512] pnew ++ [19] new_count

  unsigned long long* cand_g = (unsigned long long*)d_ws;  // 19*16*32 u64 = 76 KB

  dim3 gridA(SPLITS, NCLS);
  proto_topk_kernel<<<gridA, TA, 0, stream>>>(cam, seg, dmask, cand_g);
  proto_agg_kernel<<<NCLS, 32, 0, stream>>>(v, proto, ucount, cand_g, out);
}